// SpikeICSPBLMMinimal_35373350650277
// MI455X (gfx1250) — compile-verified
//
#include <hip/hip_runtime.h>

// Problem constants (match reference)
#define HD   192   // hidden dim
#define SL   128   // seq len
#define BT   512   // batch
#define VS   256   // vocab
#define PS   64    // patch slots
#define PHS  16    // phase dim

typedef __attribute__((ext_vector_type(2))) float v2f;
typedef __attribute__((ext_vector_type(8))) float v8f;
typedef __attribute__((ext_vector_type(4))) int   v4i;

enum { ACT_NONE = 0, ACT_GELU = 1, ACT_TANH = 2 };

#if defined(__gfx1250__) && __has_builtin(__builtin_amdgcn_global_load_async_to_lds_b128)
#define HAVE_ASYNC_LDS 1
#else
#define HAVE_ASYNC_LDS 0
#endif

__device__ __forceinline__ float gelu_exact(float x) {
  // exact erf GELU: 0.5*x*(1+erf(x/sqrt(2)))
  return 0.5f * x * (1.0f + erff(x * 0.70710678118654752440f));
}

// butterfly reductions across an aligned group of 8 lanes (wave32)
__device__ __forceinline__ float grp8_sum(float v) {
#pragma unroll
  for (int o = 1; o < 8; o <<= 1) v += __shfl_xor(v, o, 8);
  return v;
}
__device__ __forceinline__ float grp8_max(float v) {
#pragma unroll
  for (int o = 1; o < 8; o <<= 1) v = fmaxf(v, __shfl_xor(v, o, 8));
  return v;
}

// ---- WMMA f32 16x16x4 helpers -------------------------------------------
// A (16x4): lane l holds rows M=l&15, K = k0 + v + 2*(l>>4)   (v = vgpr 0/1)
// B (4x16): lane l holds col N=n0+(l&15), K = k0 + v + 2*(l>>4)
// C/D (16x16): vgpr r -> row r + 8*(l>>4), col n0 + (l&15)

template <int NT>
__device__ __forceinline__ void gemm_wmma(v8f* acc, const int* n0,
                                          const float* A, int lda, int K,
                                          const float* B, int ldb, int lane) {
  const int r  = lane & 15;          // A row index == B column offset
  const int kh = (lane >> 4) << 1;   // per-lane-half K offset (0 or 2)
  for (int k0 = 0; k0 < K; k0 += 4) {
    v2f a;
    a.x = A[r * lda + k0 + kh];
    a.y = A[r * lda + k0 + kh + 1];
#pragma unroll
    for (int t = 0; t < NT; ++t) {
      const float* Bp = B + (k0 + kh) * ldb + n0[t] + r;
      v2f b;
      b.x = Bp[0];
      b.y = Bp[ldb];
      acc[t] = __builtin_amdgcn_wmma_f32_16x16x4_f32(false, a, false, b,
                                                     (short)0, acc[t],
                                                     false, false);
    }
  }
}

__device__ __forceinline__ void store_tile(float* dst, int ldd, int n0,
                                           const v8f& d, int lane) {
  const int col = n0 + (lane & 15);
  const int rb  = (lane >> 4) << 3;
#pragma unroll
  for (int r = 0; r < 8; ++r) dst[(rb + r) * ldd + col] = d[r];
}

__device__ __forceinline__ void store_bias_act(float* dst, int ldd, int n0,
                                               const v8f& d, const float* bias,
                                               int act, int lane) {
  const int col = n0 + (lane & 15);
  const int rb  = (lane >> 4) << 3;
  const float bv = bias[col];
#pragma unroll
  for (int r = 0; r < 8; ++r) {
    float v = d[r] + bv;
    if (act == ACT_GELU)      v = gelu_exact(v);
    else if (act == ACT_TANH) v = tanhf(v);
    dst[(rb + r) * ldd + col] = v;
  }
}

// ---- Kernel 1: recurrent scan. 32 blocks x 128 threads (4 waves).
// Each block owns 16 batch rows; state kept in LDS across all 128 steps.
__global__ __launch_bounds__(128)
void rec_scan_kernel(const int* __restrict__ ids, const float* __restrict__ emb,
                     const float* __restrict__ selW, const float* __restrict__ selb,
                     const float* __restrict__ patchV,
                     const float* __restrict__ bW1, const float* __restrict__ bb1,
                     const float* __restrict__ bW2, const float* __restrict__ bb2,
                     const float* __restrict__ phaseE,
                     const float* __restrict__ rW,  const float* __restrict__ rb,
                     const float* __restrict__ sW1, const float* __restrict__ sb1,
                     const float* __restrict__ sW2, const float* __restrict__ sb2,
                     const float* __restrict__ gW1, const float* __restrict__ gb1,
                     const float* __restrict__ gW2, const float* __restrict__ gb2,
                     const float* __restrict__ lng, const float* __restrict__ lnb,
                     float* __restrict__ out_hidden, float* __restrict__ out_pw,
                     float* __restrict__ out_phw,    float* __restrict__ out_gate) {
  // Buffer lifetimes are disjoint where aliased:
  __shared__ __align__(16) float bufA[16 * HD];  // tok, later phs
  __shared__ __align__(16) float bufB[16 * HD];  // patch
  __shared__ __align__(16) float bufC[16 * HD];  // prev / hidden (state)
  __shared__ __align__(16) float bufD[16 * HD];  // sec, later succ
  __shared__ __align__(16) float bufE[16 * HD];  // h1 scratch; pw/phw raw

  const int tid  = threadIdx.x;
  const int lane = tid & 31;
  const int w    = tid >> 5;         // wave id 0..3
  const int b0   = blockIdx.x * 16;  // batch tile base
  const int row8 = tid >> 3;         // row owned in 8-lane-group ops (0..15)
  const int sub8 = tid & 7;          // lane within the 8-lane group
  int n0s[3] = { w * 16, (w + 4) * 16, (w + 8) * 16 };  // 12 n-tiles / 4 waves

  for (int i = tid; i < 16 * HD; i += 128) bufC[i] = 0.0f;  // prev0 = 0
  __syncthreads();

#pragma unroll 1
  for (int s = 0; s < SL; ++s) {
    // (1) token embedding gather -> bufA (16 rows x 192 f32 = 12 KB)
#if HAVE_ASYNC_LDS
    {
      // 768 x 16B transfers, 6 per thread, direct global -> LDS (ASYNCcnt)
#pragma unroll
      for (int it = 0; it < 6; ++it) {
        const int i = tid + it * 128;           // 0..767
        const int m = i / 48, c = i - m * 48;   // row, 16B-chunk in row
        const int id = ids[(b0 + m) * SL + s];
        const float* gp = emb + (size_t)id * HD + c * 4;
        __builtin_amdgcn_global_load_async_to_lds_b128(
            (__attribute__((address_space(1))) v4i*)gp,
            (__attribute__((address_space(3))) v4i*)(bufA + m * HD + c * 4),
            0, 0);
      }
#if __has_builtin(__builtin_amdgcn_s_wait_asynccnt)
      __builtin_amdgcn_s_wait_asynccnt(0);
#else
      asm volatile("s_wait_asynccnt 0x0" ::: "memory");
#endif
    }
#else
    for (int i = tid; i < 16 * HD; i += 128) {
      const int m = i / HD, k = i - m * HD;
      bufA[i] = emb[ids[(b0 + m) * SL + s] * HD + k];
    }
#endif
    // prefetch next step's embedding rows while this step computes
    if (tid < 16 && s + 1 < SL)
      __builtin_prefetch(emb + (size_t)ids[(b0 + tid) * SL + s + 1] * HD, 0, 1);
    __syncthreads();

    // (2) selector logits: [tok|prev] @ selW(384x64) + selb -> bufE[16x64]
    {
      v8f acc = {};
      int n0 = w * 16;  // 4 n-tiles, one per wave
      gemm_wmma<1>(&acc, &n0, bufA, HD, HD, selW, PS, lane);
      gemm_wmma<1>(&acc, &n0, bufC, HD, HD, selW + HD * PS, PS, lane);
      store_bias_act(bufE, PS, n0, acc, selb, ACT_NONE, lane);
    }
    __syncthreads();
    // softmax over 64, 8 lanes/row x 8 elems
    {
      float* rp = bufE + row8 * PS;
      float e[8], mx = -3.402823466e+38f;
#pragma unroll
      for (int j = 0; j < 8; ++j) { e[j] = rp[sub8 * 8 + j]; mx = fmaxf(mx, e[j]); }
      mx = grp8_max(mx);
      float sum = 0.0f;
#pragma unroll
      for (int j = 0; j < 8; ++j) { e[j] = expf(e[j] - mx); sum += e[j]; }
      const float inv = 1.0f / grp8_sum(sum);
      float* op = out_pw + ((size_t)(b0 + row8) * SL + s) * PS;
#pragma unroll
      for (int j = 0; j < 8; ++j) {
        const float v = e[j] * inv;
        rp[sub8 * 8 + j] = v;
        op[sub8 * 8 + j] = v;
      }
    }
    __syncthreads();

    // (3) patch = pw @ patch_values(64x192) -> bufB
    {
      v8f acc[3] = {};
      gemm_wmma<3>(acc, n0s, bufE, PS, PS, patchV, HD, lane);
#pragma unroll
      for (int t = 0; t < 3; ++t) store_tile(bufB, HD, n0s[t], acc[t], lane);
    }
    __syncthreads();

    // (4) bind layer 1: [tok|patch|prev] @ bW1(576x192) + bb1, gelu -> bufE
    {
      v8f acc[3] = {};
      gemm_wmma<3>(acc, n0s, bufA, HD, HD, bW1,               HD, lane);
      gemm_wmma<3>(acc, n0s, bufB, HD, HD, bW1 + HD * HD,     HD, lane);
      gemm_wmma<3>(acc, n0s, bufC, HD, HD, bW1 + 2 * HD * HD, HD, lane);
#pragma unroll
      for (int t = 0; t < 3; ++t) store_bias_act(bufE, HD, n0s[t], acc[t], bb1, ACT_GELU, lane);
    }
    __syncthreads();

    // (5) bind layer 2: h1 @ bW2(192x192) + bb2, tanh -> bufD (sec)
    {
      v8f acc[3] = {};
      gemm_wmma<3>(acc, n0s, bufE, HD, HD, bW2, HD, lane);
#pragma unroll
      for (int t = 0; t < 3; ++t) store_bias_act(bufD, HD, n0s[t], acc[t], bb2, ACT_TANH, lane);
    }
    __syncthreads();

    // (6) router: [sec|prev] @ rW(384x16) + rb -> bufE[16x16] (wave 0)
    if (w == 0) {
      v8f acc = {};
      int n0 = 0;
      gemm_wmma<1>(&acc, &n0, bufD, HD, HD, rW, PHS, lane);
      gemm_wmma<1>(&acc, &n0, bufC, HD, HD, rW + HD * PHS, PHS, lane);
      store_bias_act(bufE, PHS, 0, acc, rb, ACT_NONE, lane);
    }
    __syncthreads();
    // softmax over 16, 8 lanes/row x 2 elems
    {
      float* rp = bufE + row8 * PHS;
      float e0 = rp[sub8 * 2], e1 = rp[sub8 * 2 + 1];
      const float mx = grp8_max(fmaxf(e0, e1));
      e0 = expf(e0 - mx); e1 = expf(e1 - mx);
      const float inv = 1.0f / grp8_sum(e0 + e1);
      e0 *= inv; e1 *= inv;
      rp[sub8 * 2] = e0; rp[sub8 * 2 + 1] = e1;
      float* op = out_phw + ((size_t)(b0 + row8) * SL + s) * PHS;
      op[sub8 * 2] = e0; op[sub8 * 2 + 1] = e1;
    }
    __syncthreads();

    // (7) phs = phw @ phase_embed(16x192) -> bufA (tok is dead)
    {
      v8f acc[3] = {};
      gemm_wmma<3>(acc, n0s, bufE, PHS, PHS, phaseE, HD, lane);
#pragma unroll
      for (int t = 0; t < 3; ++t) store_tile(bufA, HD, n0s[t], acc[t], lane);
    }
    __syncthreads();

    // (8) succ layer 1: [sec|patch|phs] @ sW1(576x192) + sb1, gelu -> bufE
    {
      v8f acc[3] = {};
      gemm_wmma<3>(acc, n0s, bufD, HD, HD, sW1,               HD, lane);
      gemm_wmma<3>(acc, n0s, bufB, HD, HD, sW1 + HD * HD,     HD, lane);
      gemm_wmma<3>(acc, n0s, bufA, HD, HD, sW1 + 2 * HD * HD, HD, lane);
#pragma unroll
      for (int t = 0; t < 3; ++t) store_bias_act(bufE, HD, n0s[t], acc[t], sb1, ACT_GELU, lane);
    }
    __syncthreads();

    // (9) succ layer 2: h1 @ sW2 + sb2, tanh -> bufD (sec is dead)
    {
      v8f acc[3] = {};
      gemm_wmma<3>(acc, n0s, bufE, HD, HD, sW2, HD, lane);
#pragma unroll
      for (int t = 0; t < 3; ++t) store_bias_act(bufD, HD, n0s[t], acc[t], sb2, ACT_TANH, lane);
    }
    __syncthreads();

    // (10) gate layer 1: [succ|prev] @ gW1(384x192) + gb1, gelu -> bufE
    {
      v8f acc[3] = {};
      gemm_wmma<3>(acc, n0s, bufD, HD, HD, gW1,           HD, lane);
      gemm_wmma<3>(acc, n0s, bufC, HD, HD, gW1 + HD * HD, HD, lane);
#pragma unroll
      for (int t = 0; t < 3; ++t) store_bias_act(bufE, HD, n0s[t], acc[t], gb1, ACT_GELU, lane);
    }
    __syncthreads();

    // (11) gate scalar + gated blend + LayerNorm, 8 lanes/row x 24 elems
    {
      const int k0 = sub8 * 24;
      float* h1r   = bufE + row8 * HD;
      float* prevr = bufC + row8 * HD;
      float* succr = bufD + row8 * HD;

      float pa = 0.0f;
#pragma unroll
      for (int k = 0; k < 24; ++k) pa += h1r[k0 + k] * gW2[k0 + k];
      const float g = 1.0f / (1.0f + expf(-(grp8_sum(pa) + gb2[0])));
      if (sub8 == 0) out_gate[(size_t)(b0 + row8) * SL + s] = g;

      float hv[24], pm = 0.0f;
#pragma unroll
      for (int k = 0; k < 24; ++k) {
        hv[k] = g * succr[k0 + k] + (1.0f - g) * prevr[k0 + k];
        pm += hv[k];
      }
      const float mean = grp8_sum(pm) * (1.0f / HD);
      float pv = 0.0f;
#pragma unroll
      for (int k = 0; k < 24; ++k) { const float d = hv[k] - mean; pv += d * d; }
      const float inv = rsqrtf(grp8_sum(pv) * (1.0f / HD) + 1e-5f);
      float* oh = out_hidden + ((size_t)(b0 + row8) * SL + s) * HD;
#pragma unroll
      for (int k = 0; k < 24; ++k) {
        const float y = (hv[k] - mean) * inv * lng[k0 + k] + lnb[k0 + k];
        prevr[k0 + k] = y;   // becomes prev for next step
        oh[k0 + k] = y;
      }
    }
    __syncthreads();
  }
}

// ---- Kernel 2: decoder GEMM, hoisted out of the scan (fully parallel).
// logits[B*S,256] = hidden[B*S,192] @ dec_W[192,256]
__global__ __launch_bounds__(128)
void dec_kernel(const float* __restrict__ hidden, const float* __restrict__ decW,
                float* __restrict__ logits) {
  const int lane = threadIdx.x & 31;
  const int w    = threadIdx.x >> 5;
  const size_t rt = blockIdx.x;               // row tile (16 rows)
  const float* A = hidden + rt * 16 * HD;
  float* out     = logits + rt * 16 * VS;

  int n0s[4] = { (w * 4 + 0) * 16, (w * 4 + 1) * 16, (w * 4 + 2) * 16, (w * 4 + 3) * 16 };
  v8f acc[4] = {};
  gemm_wmma<4>(acc, n0s, A, HD, HD, decW, VS, lane);
#pragma unroll
  for (int t = 0; t < 4; ++t) store_tile(out, VS, n0s[t], acc[t], lane);
}

extern "C" void kernel_launch(void* const* d_in, const int* in_sizes, int n_in,
                              void* d_out, int out_size, void* d_ws, size_t ws_size,
                              hipStream_t stream) {
  (void)in_sizes; (void)n_in; (void)out_size; (void)d_ws; (void)ws_size;
  const int*   ids    = (const int*)d_in[0];
  const float* emb    = (const float*)d_in[1];
  const float* selW   = (const float*)d_in[2];
  const float* selb   = (const float*)d_in[3];
  const float* patchV = (const float*)d_in[4];
  const float* bW1    = (const float*)d_in[5];
  const float* bb1    = (const float*)d_in[6];
  const float* bW2    = (const float*)d_in[7];
  const float* bb2    = (const float*)d_in[8];
  const float* phaseE = (const float*)d_in[9];
  const float* rW     = (const float*)d_in[10];
  const float* rb     = (const float*)d_in[11];
  const float* sW1    = (const float*)d_in[12];
  const float* sb1    = (const float*)d_in[13];
  const float* sW2    = (const float*)d_in[14];
  const float* sb2    = (const float*)d_in[15];
  const float* gW1    = (const float*)d_in[16];
  const float* gb1    = (const float*)d_in[17];
  const float* gW2    = (const float*)d_in[18];
  const float* gb2    = (const float*)d_in[19];
  const float* lng    = (const float*)d_in[20];
  const float* lnb    = (const float*)d_in[21];
  const float* decW   = (const float*)d_in[22];

  float* out = (float*)d_out;
  float* o_logits = out;                                   // [B,S,V] 16,777,216
  float* o_hidden = out + (size_t)16777216;                // [B,S,H] 12,582,912
  float* o_pw     = out + (size_t)29360128;                // [B,S,P]  4,194,304
  float* o_phw    = out + (size_t)33554432;                // [B,S,PH] 1,048,576
  float* o_gate   = out + (size_t)34603008;                // [B,S,1]     65,536

  rec_scan_kernel<<<BT / 16, 128, 0, stream>>>(
      ids, emb, selW, selb, patchV, bW1, bb1, bW2, bb2, phaseE, rW, rb,
      sW1, sb1, sW2, sb2, gW1, gb1, gW2, gb2, lng, lnb,
      o_hidden, o_pw, o_phw, o_gate);

  dec_kernel<<<(BT * SL) / 16, 128, 0, stream>>>(o_hidden, decW, o_logits);
}